// BaseDenseAttention_17386027614705
// MI455X (gfx1250) — compile-verified
//
#include <hip/hip_runtime.h>
#include <hip/hip_bf16.h>
#include <cstdint>

typedef __attribute__((ext_vector_type(2))) float v2f;
typedef __attribute__((ext_vector_type(8))) float v8f;

#define B_ 8
#define T_ 2048
#define D_ 64

__device__ __forceinline__ v8f wmma_f32(v2f a, v2f b, v8f c) {
  // D = A(16x4,f32) * B(4x16,f32) + C(16x16,f32)  -> v_wmma_f32_16x16x4_f32
  return __builtin_amdgcn_wmma_f32_16x16x4_f32(false, a, false, b, (short)0, c,
                                               false, false);
}

__global__ __launch_bounds__(128) void attn_fp32_wmma(
    const float* __restrict__ q, const float* __restrict__ v,
    const float* __restrict__ k, const unsigned char* __restrict__ qmask,
    const unsigned char* __restrict__ vmask, float* __restrict__ out) {
  // per-wave 16x16 staging tile (C-layout -> A-layout relayout), padded
  __shared__ float lds[4][16][20];

  const int lane = threadIdx.x & 31;
  const int wv   = threadIdx.x >> 5;
  const int gw   = blockIdx.x * 4 + wv;   // global wave id: 0..1023
  const int b    = gw >> 7;               // 128 q-blocks per batch
  const int qb   = gw & 127;              // 16-row query block index

  const int half = lane >> 4;             // 0: lanes 0-15, 1: lanes 16-31
  const int nn   = lane & 15;             // col (B/C/D) or row (A) index
  const int koff = half * 2;              // A/B K-offset for upper half-wave

  const size_t rowBase = (size_t)(b * T_ + qb * 16);  // first query row
  float* wout = out;                                   // weights [B,T,T]
  float* rout = out + (size_t)B_ * T_ * T_;            // result  [B,T,D]

  // ---- load Q fragments (A-layout): 16 fragments cover D=64 (K=4 each) ----
  v2f qf[16];
  {
    const float* qbase = q + (rowBase + (size_t)nn) * D_;
#pragma unroll
    for (int f = 0; f < 16; ++f) {
      const float* p = qbase + 4 * f + koff;
      qf[f].x = p[0];
      qf[f].y = p[1];
    }
  }

  // per-lane online-softmax stats for rows (vv + 8*half), vv = 0..7
  float runm[8], runs[8];
#pragma unroll
  for (int i = 0; i < 8; ++i) { runm[i] = -3.0e38f; runs[i] = 0.0f; }

  const int KT = qb + 1;  // causal: number of 16-wide key tiles

  // ================= pass 1: row max / row sum(exp) =================
  for (int kt = 0; kt < KT; ++kt) {
    v8f c = {};
    const float* kbase = k + ((size_t)(b * T_ + kt * 16) + (size_t)nn) * D_;
#pragma unroll
    for (int f = 0; f < 16; ++f) {
      v2f kf;
      const float* p = kbase + 4 * f + koff;
      kf.x = p[0];
      kf.y = p[1];
      c = wmma_f32(qf[f], kf, c);  // scores += Q[:,4f:4f+4] * K[:,4f:4f+4]^T
    }
    const bool diag = (kt == qb);
    const bool vm = vmask[(size_t)b * T_ + kt * 16 + nn] != 0;
#pragma unroll
    for (int vv = 0; vv < 8; ++vv) {
      const int mrow = vv + 8 * half;
      const bool ok = vm && !(diag && nn > mrow);
      float sc = ok ? c[vv] : c[vv] - 1e9f;
      // reduce max over the 16 cols of this row (within 16-lane half)
      float tm = sc;
#pragma unroll
      for (int m = 1; m < 16; m <<= 1) tm = fmaxf(tm, __shfl_xor(tm, m, 32));
      const float nm = fmaxf(runm[vv], tm);
      float e = __expf(sc - nm);
#pragma unroll
      for (int m = 1; m < 16; m <<= 1) e += __shfl_xor(e, m, 32);
      runs[vv] = runs[vv] * __expf(runm[vv] - nm) + e;
      runm[vv] = nm;
    }
  }

  float rowri[8];
#pragma unroll
  for (int vv = 0; vv < 8; ++vv) rowri[vv] = 1.0f / runs[vv];

  // ============ pass 2: weights out + W @ V accumulation ============
  v8f acc0 = {}, acc1 = {}, acc2 = {}, acc3 = {};
  for (int kt = 0; kt < KT; ++kt) {
    v8f c = {};
    const float* kbase = k + ((size_t)(b * T_ + kt * 16) + (size_t)nn) * D_;
#pragma unroll
    for (int f = 0; f < 16; ++f) {
      v2f kf;
      const float* p = kbase + 4 * f + koff;
      kf.x = p[0];
      kf.y = p[1];
      c = wmma_f32(qf[f], kf, c);
    }
    const bool diag = (kt == qb);
    const bool vm = vmask[(size_t)b * T_ + kt * 16 + nn] != 0;
#pragma unroll
    for (int vv = 0; vv < 8; ++vv) {
      const int mrow = vv + 8 * half;
      const bool ok = vm && !(diag && nn > mrow);
      const float sc = ok ? c[vv] : c[vv] - 1e9f;
      const float w = __expf(sc - runm[vv]);       // unnormalized weight
      lds[wv][mrow][nn] = w;                       // stage for A-layout reads
      wout[(rowBase + (size_t)mrow) * T_ + kt * 16 + nn] = w * rowri[vv];
    }
    // W(16x16, from LDS) @ V(16x64): contract 4 keys per wmma, 4 n-tiles
#pragma unroll
    for (int ks = 0; ks < 4; ++ks) {
      v2f a;  // A-layout: row = nn, K = 4*ks + koff .. +1
      a.x = lds[wv][nn][4 * ks + koff];
      a.y = lds[wv][nn][4 * ks + koff + 1];
      const float* vb =
          v + ((size_t)(b * T_ + kt * 16 + 4 * ks + koff)) * D_;
      {
        v2f bf; bf.x = vb[0 * 16 + nn]; bf.y = vb[D_ + 0 * 16 + nn];
        acc0 = wmma_f32(a, bf, acc0);
      }
      {
        v2f bf; bf.x = vb[1 * 16 + nn]; bf.y = vb[D_ + 1 * 16 + nn];
        acc1 = wmma_f32(a, bf, acc1);
      }
      {
        v2f bf; bf.x = vb[2 * 16 + nn]; bf.y = vb[D_ + 2 * 16 + nn];
        acc2 = wmma_f32(a, bf, acc2);
      }
      {
        v2f bf; bf.x = vb[3 * 16 + nn]; bf.y = vb[D_ + 3 * 16 + nn];
        acc3 = wmma_f32(a, bf, acc3);
      }
    }
  }

  // ---- zero-fill strictly-upper (non-causal) weight columns ----
  {
    const int zstart = KT * 16;  // first zero column (multiple of 16)
    const float4 z4 = make_float4(0.f, 0.f, 0.f, 0.f);
    for (int r = 0; r < 16; ++r) {
      float* rowp = wout + (rowBase + (size_t)r) * T_;
      for (int cc = zstart + 4 * lane; cc < T_; cc += 128)
        *(float4*)(rowp + cc) = z4;
    }
  }

  // ---- write result: normalize rows, apply q_mask ----
#pragma unroll
  for (int vv = 0; vv < 8; ++vv) {
    const int mrow = vv + 8 * half;
    const float qm = (qmask[rowBase + (size_t)mrow] != 0) ? 1.0f : 0.0f;
    const float s = rowri[vv] * qm;
    float* rp = rout + (rowBase + (size_t)mrow) * D_ + nn;
    rp[0]  = acc0[vv] * s;
    rp[16] = acc1[vv] * s;
    rp[32] = acc2[vv] * s;
    rp[48] = acc3[vv] * s;
  }
}

extern "C" void kernel_launch(void* const* d_in, const int* in_sizes, int n_in,
                              void* d_out, int out_size, void* d_ws,
                              size_t ws_size, hipStream_t stream) {
  // setup_inputs() order: q, v, k, q_mask, v_mask
  const float* q = (const float*)d_in[0];
  const float* v = (const float*)d_in[1];
  const float* k = (const float*)d_in[2];
  const unsigned char* qm = (const unsigned char*)d_in[3];
  const unsigned char* vm = (const unsigned char*)d_in[4];
  float* out = (float*)d_out;

  // 1024 query blocks (B * T/16), 4 waves (= 4 blocks' worth) per workgroup
  dim3 grid(256), block(128);
  attn_fp32_wmma<<<grid, block, 0, stream>>>(q, v, k, qm, vm, out);
}